// _GNNEncoder_73890617360784
// MI455X (gfx1250) — compile-verified
//
#include <hip/hip_runtime.h>
#include <stdint.h>

#define N_NODES 50000
#define N_EDGES 800000
#define DIM     256
#define LAYERS  3
#define GRAPHS  512
#define LD_OUT  (LAYERS*DIM)   // 768
#define BK      16             // K-tile (double-buffered async staging)
#define NSTAGE  (DIM/BK)       // 16

typedef float v2f __attribute__((ext_vector_type(2)));
typedef float v8f __attribute__((ext_vector_type(8)));

// ---------------- utility kernels ----------------

__global__ void gnn_zero(float* p, long n) {
    long i = (long)blockIdx.x * blockDim.x + threadIdx.x;
    if (i < n) p[i] = 0.f;
}

__global__ void gnn_counts(const int* __restrict__ batch, float* __restrict__ counts) {
    int i = blockIdx.x * blockDim.x + threadIdx.x;
    if (i < N_NODES) atomicAdd(&counts[batch[i]], 1.f);
}

// agg[n][:] = h[n][:]   (h may be strided: ldh = 256 for x, 768 for node_embed slices)
__global__ void gnn_copy_h(const float* __restrict__ h, int ldh, float* __restrict__ agg) {
    long i  = (long)blockIdx.x * blockDim.x + threadIdx.x;  // one float4 per thread
    long n  = i >> 6;                                       // 64 float4 per row
    int  c4 = (int)(i & 63);
    float4 v = *((const float4*)(h + n * (long)ldh) + c4);
    *((float4*)(agg + n * (long)DIM) + c4) = v;
}

// agg[dst[e]][:] += h[src[e]][:]   64 threads per edge, 4 floats per thread
__global__ void gnn_edge_scatter(const int* __restrict__ srcv, const int* __restrict__ dstv,
                                 const float* __restrict__ h, int ldh,
                                 float* __restrict__ agg) {
    long t = (long)blockIdx.x * blockDim.x + threadIdx.x;
    long e = t >> 6;
    int  q = (int)(t & 63);
    int  s = srcv[e];
    int  d = dstv[e];
    float4 v = *((const float4*)(h + (long)s * ldh) + q);
    float* out = agg + (long)d * DIM + (q << 2);
    atomicAdd(out + 0, v.x);
    atomicAdd(out + 1, v.y);
    atomicAdd(out + 2, v.z);
    atomicAdd(out + 3, v.w);
}

// Issue one K-stage (BK x 256 floats = 16KB) of W into LDS via async DMA.
// 1024 16-byte chunks; 32 wave-instructions per block -> exactly 4 per wave32.
// All addresses are 16B aligned (W row = 1KB, LDS row pitch = 260 floats = 1040B).
__device__ __forceinline__ void gnn_issue_b_async(const float* W, int k0, unsigned bsbase,
                                                  int wave, int lane) {
    #pragma unroll
    for (int r = 0; r < 4; ++r) {
        int chunk = (r * 8 + wave) * 32 + lane;          // 0..1023
        int row = chunk >> 6;                            // 0..15
        int c4  = chunk & 63;                            // 0..63
        unsigned long long g =
            (unsigned long long)(uintptr_t)(W + (size_t)(k0 + row) * DIM + c4 * 4);
        unsigned l = bsbase + (unsigned)((row * (DIM + 4) + c4 * 4) * 4);
        asm volatile("global_load_async_to_lds_b128 %0, %1, off"
                     :: "v"(l), "v"(g) : "memory");
    }
}

// ---------------- WMMA GEMM: C[m, :] = relu(A[m, :] @ W + bias) ----------------
// A: [N_NODES, 256] contiguous. W: [256,256] row-major. C: ldc-strided.
// Block owns 16 full rows (all 256 cols) -> safe for in-place C==A across blocks.
// 8 wave32 per block; wave w computes cols [32w, 32w+32) as two 16x16 tiles.
// W tiles are double-buffered in LDS via GLOBAL_LOAD_ASYNC_TO_LDS_B128 (ASYNCcnt).
__global__ __launch_bounds__(256) void gnn_gemm_bias_relu(
    const float* A, const float* __restrict__ W,
    const float* __restrict__ bias, float* C, int ldc) {
    __shared__ __align__(16) float As[16][BK + 4];
    __shared__ __align__(16) float Bs[2][BK][DIM + 4];

    const int m0   = blockIdx.x * 16;
    const int tid  = threadIdx.x;
    const int lane = tid & 31;
    const int wave = tid >> 5;     // 0..7
    const int l16  = lane & 15;
    const int half = lane >> 4;    // 0: K-pair {0,1} / M 0-7 ; 1: K-pair {2,3} / M 8-15

    const unsigned bs0 = (unsigned)(uintptr_t)&Bs[0][0][0];
    const unsigned bs1 = (unsigned)(uintptr_t)&Bs[1][0][0];

    v8f acc0 = {};
    v8f acc1 = {};

    // prologue: async-stage W K-tile 0 into buffer 0
    gnn_issue_b_async(W, 0, bs0, wave, lane);

    for (int s = 0; s < NSTAGE; ++s) {
        const int k0  = s * BK;
        const int cur = s & 1;

        // kick off DMA for the next K-tile into the other buffer
        if (s < NSTAGE - 1)
            gnn_issue_b_async(W, k0 + BK, cur ? bs0 : bs1, wave, lane);

        // synchronous A tile stage (16 x 16 floats = 64 float4), issued before the
        // async wait so its global latency overlaps
        const int arow = tid >> 2, ac4 = tid & 3;
        float4 av;
        if (tid < 64)
            av = *((const float4*)(A + (long)(m0 + arow) * DIM + k0) + ac4);

        // wait until the *current* tile's DMA has landed (async loads retire in
        // order: <=4 outstanding means only the just-issued next-stage ops remain)
        if (s < NSTAGE - 1) asm volatile("s_wait_asynccnt 0x4" ::: "memory");
        else                asm volatile("s_wait_asynccnt 0x0" ::: "memory");

        if (tid < 64) {
            As[arow][ac4 * 4 + 0] = av.x; As[arow][ac4 * 4 + 1] = av.y;
            As[arow][ac4 * 4 + 2] = av.z; As[arow][ac4 * 4 + 3] = av.w;
        }
        __syncthreads();

        const float (*Bc)[DIM + 4] = Bs[cur];
        #pragma unroll
        for (int kk = 0; kk < BK; kk += 4) {
            const int kb = kk + half * 2;   // ISA 32-bit A layout: VGPR0=K{0,2}, VGPR1=K{1,3}
            v2f a;
            a[0] = As[l16][kb];
            a[1] = As[l16][kb + 1];
            v2f b0, b1;
            b0[0] = Bc[kb][wave * 32 + l16];
            b0[1] = Bc[kb + 1][wave * 32 + l16];
            b1[0] = Bc[kb][wave * 32 + 16 + l16];
            b1[1] = Bc[kb + 1][wave * 32 + 16 + l16];
            acc0 = __builtin_amdgcn_wmma_f32_16x16x4_f32(false, a, false, b0,
                                                         (short)0, acc0, false, false);
            acc1 = __builtin_amdgcn_wmma_f32_16x16x4_f32(false, a, false, b1,
                                                         (short)0, acc1, false, false);
        }
        __syncthreads();
    }

    // epilogue: bias (constant per lane column) + ReLU + strided store
    const int col0 = wave * 32 + l16;
    const int col1 = col0 + 16;
    const float bb0 = bias[col0];
    const float bb1 = bias[col1];
    #pragma unroll
    for (int v = 0; v < 8; v++) {
        const long row = m0 + half * 8 + v;   // C/D layout: VGPR v -> M = v (+8 for hi half)
        float r0 = acc0[v] + bb0; r0 = r0 > 0.f ? r0 : 0.f;
        float r1 = acc1[v] + bb1; r1 = r1 > 0.f ? r1 : 0.f;
        C[row * (long)ldc + col0] = r0;
        C[row * (long)ldc + col1] = r1;
    }
}

// ---------------- pooling ----------------

__global__ void gnn_pool_accum(const float* __restrict__ h, int ldh,
                               const int* __restrict__ batch,
                               float* __restrict__ g, int ldg) {
    long i = (long)blockIdx.x * blockDim.x + threadIdx.x;
    long n = i >> 8;            // DIM = 256 cols
    int  c = (int)(i & 255);
    int  gi = batch[n];
    atomicAdd(g + (long)gi * ldg + c, h[n * (long)ldh + c]);
}

__global__ void gnn_pool_div(float* __restrict__ graph, const float* __restrict__ counts) {
    int i = blockIdx.x * blockDim.x + threadIdx.x;   // exactly GRAPHS*LD_OUT threads
    int g = i / LD_OUT;
    float c = counts[g];
    c = c < 1.f ? 1.f : c;
    graph[i] *= (1.f / c);
}

// ---------------- launch ----------------

extern "C" void kernel_launch(void* const* d_in, const int* in_sizes, int n_in,
                              void* d_out, int out_size, void* d_ws, size_t ws_size,
                              hipStream_t stream) {
    (void)in_sizes; (void)n_in; (void)out_size; (void)ws_size;

    const float* x     = (const float*)d_in[0];
    const int*   ei    = (const int*)d_in[1];
    const int*   batch = (const int*)d_in[2];
    const float* Ws1   = (const float*)d_in[3];
    const float* bs1   = (const float*)d_in[4];
    const float* Ws2   = (const float*)d_in[5];
    const float* bs2   = (const float*)d_in[6];

    float* out   = (float*)d_out;
    float* graph = out;                                  // [GRAPHS, 768]
    float* node  = out + (size_t)GRAPHS * LD_OUT;        // [N_NODES, 768]

    float* agg    = (float*)d_ws;                        // [N_NODES, 256]
    float* counts = agg + (size_t)N_NODES * DIM;         // [GRAPHS]

    const int* srcv = ei;              // edge_index[0]
    const int* dstv = ei + N_EDGES;    // edge_index[1]

    gnn_zero<<<(GRAPHS * LD_OUT + 255) / 256, 256, 0, stream>>>(graph, (long)GRAPHS * LD_OUT);
    gnn_zero<<<(GRAPHS + 255) / 256, 256, 0, stream>>>(counts, GRAPHS);
    gnn_counts<<<(N_NODES + 255) / 256, 256, 0, stream>>>(batch, counts);

    for (int i = 0; i < LAYERS; i++) {
        const float* h = (i == 0) ? x : (node + (size_t)(i - 1) * DIM);
        const int ldh  = (i == 0) ? DIM : LD_OUT;

        gnn_copy_h<<<(N_NODES * (DIM / 4)) / 256, 256, 0, stream>>>(h, ldh, agg);
        gnn_edge_scatter<<<(N_EDGES * 64) / 256, 256, 0, stream>>>(srcv, dstv, h, ldh, agg);

        // t = relu(agg @ W1 + b1)  — in place (blocks own full rows)
        gnn_gemm_bias_relu<<<N_NODES / 16, 256, 0, stream>>>(
            agg, Ws1 + (size_t)i * DIM * DIM, bs1 + (size_t)i * DIM, agg, DIM);
        // h = relu(t @ W2 + b2)  — written straight into node_embed slice
        gnn_gemm_bias_relu<<<N_NODES / 16, 256, 0, stream>>>(
            agg, Ws2 + (size_t)i * DIM * DIM, bs2 + (size_t)i * DIM,
            node + (size_t)i * DIM, LD_OUT);

        gnn_pool_accum<<<(N_NODES * DIM) / 256, 256, 0, stream>>>(
            node + (size_t)i * DIM, LD_OUT, batch, graph + (size_t)i * DIM, LD_OUT);
    }

    gnn_pool_div<<<(GRAPHS * LD_OUT) / 256, 256, 0, stream>>>(graph, counts);
}